// IsLandLoss_12678743457990
// MI455X (gfx1250) — compile-verified
//
#include <hip/hip_runtime.h>
#include <hip/hip_bf16.h>

// ---- problem constants (match reference) ----
#define NUM_CLASSES 1000
#define FEAT_DIM    512
#define BATCH       32768
#define N_PAD       1008          // 63 * 16, padded rows are zero
#define N_TILES     63            // ceil(1000/16)
#define CL_BLOCKS   4096          // center-loss partial blocks

typedef __attribute__((ext_vector_type(2))) float v2f;
typedef __attribute__((ext_vector_type(8))) float v8f;

__device__ __forceinline__ float wave_reduce(float v) {
    #pragma unroll
    for (int off = 16; off > 0; off >>= 1)
        v += __shfl_xor(v, off, 32);
    return v;
}

// ---------------------------------------------------------------------------
// Kernel 1: normalize centers -> cn[1008][512], rows >= 1000 written as zeros.
// One wave (32 lanes) per row; each lane owns 4 float4 chunks (16 floats).
// ---------------------------------------------------------------------------
__global__ __launch_bounds__(256) void k_normalize(const float* __restrict__ centers,
                                                   float* __restrict__ cn) {
    const int wave = threadIdx.x >> 5;
    const int lane = threadIdx.x & 31;
    const int row  = blockIdx.x * 8 + wave;        // 126 blocks * 8 = 1008 rows
    if (row >= N_PAD) return;

    float4 v[4];
    float ss = 0.0f;
    if (row < NUM_CLASSES) {
        const float4* src = (const float4*)(centers + row * FEAT_DIM);
        #pragma unroll
        for (int i = 0; i < 4; ++i) {
            v[i] = src[lane + 32 * i];
            ss += v[i].x * v[i].x + v[i].y * v[i].y + v[i].z * v[i].z + v[i].w * v[i].w;
        }
    } else {
        #pragma unroll
        for (int i = 0; i < 4; ++i) v[i] = make_float4(0.f, 0.f, 0.f, 0.f);
    }

    ss = wave_reduce(ss);                          // butterfly: all lanes get total
    const float inv = (row < NUM_CLASSES) ? (1.0f / fmaxf(sqrtf(ss), 1e-8f)) : 0.0f;

    float4* dst = (float4*)(cn + row * FEAT_DIM);
    #pragma unroll
    for (int i = 0; i < 4; ++i) {
        float4 o = make_float4(v[i].x * inv, v[i].y * inv, v[i].z * inv, v[i].w * inv);
        dst[lane + 32 * i] = o;
    }
}

// ---------------------------------------------------------------------------
// Kernel 2: Gram = cn @ cn^T via V_WMMA_F32_16X16X4_F32.
// One wave per 16x16 output tile; K loop of 128 WMMAs (K=512, 4 per op).
// A lane layout == B lane layout (same addressing, different row base):
//   value = cn[rowBase + (lane&15)][k + (lane>>4)*2 + vgpr]
// C layout: lane holds col = tileN + (lane&15), rows tileM + (lane>>4)*8 + i.
// Per-tile masked off-diagonal sum -> fixed-slot partial (deterministic).
// ---------------------------------------------------------------------------
__global__ __launch_bounds__(32) void k_gram_tiles(const float* __restrict__ cn,
                                                   float* __restrict__ partials) {
    const int lane = threadIdx.x;
    const int tM = blockIdx.y * 16;
    const int tN = blockIdx.x * 16;
    const int m    = lane & 15;
    const int half = lane >> 4;

    const float* arow = cn + (tM + m) * FEAT_DIM + half * 2;
    const float* brow = cn + (tN + m) * FEAT_DIM + half * 2;

    v8f c = {};
    #pragma unroll 8
    for (int k = 0; k < FEAT_DIM; k += 4) {
        v2f a = *(const v2f*)(arow + k);
        v2f b = *(const v2f*)(brow + k);
        c = __builtin_amdgcn_wmma_f32_16x16x4_f32(
                /*neg_a=*/false, a, /*neg_b=*/false, b,
                /*c_mod=*/(short)0, c, /*reuse_a=*/false, /*reuse_b=*/false);
    }

    const int col     = tN + m;
    const int rowBase = tM + half * 8;
    float s = 0.0f;
    #pragma unroll
    for (int i = 0; i < 8; ++i) {
        const int r = rowBase + i;
        // padded rows contribute exactly 0, but mask diagonal + OOB anyway
        if (r < NUM_CLASSES && col < NUM_CLASSES && r != col) s += c[i];
    }
    s = wave_reduce(s);
    if (lane == 0) partials[blockIdx.y * N_TILES + blockIdx.x] = s;
}

// ---------------------------------------------------------------------------
// Kernel 3: center-loss partial sums. Grid-stride over float4 chunks of feat
// (32768*128 chunks). One fixed-slot partial per block (deterministic).
// ---------------------------------------------------------------------------
__global__ __launch_bounds__(256) void k_center_loss(const float* __restrict__ feat,
                                                     const float* __restrict__ centers,
                                                     const int* __restrict__ label,
                                                     float* __restrict__ partials) {
    const float4* f4 = (const float4*)feat;
    const float4* c4 = (const float4*)centers;
    const int tid      = blockIdx.x * blockDim.x + threadIdx.x;
    const int nthreads = gridDim.x * blockDim.x;
    const int total    = BATCH * (FEAT_DIM / 4);   // 4,194,304 float4 chunks

    float acc = 0.0f;
    for (int i = tid; i < total; i += nthreads) {
        const int b  = i >> 7;                     // FEAT_DIM/4 = 128
        const int dv = i & 127;
        const int lab = label[b];
        const float4 f = f4[i];
        const float4 c = c4[lab * 128 + dv];
        const float dx = f.x - c.x, dy = f.y - c.y, dz = f.z - c.z, dw = f.w - c.w;
        acc += dx * dx + dy * dy + dz * dz + dw * dw;
    }

    acc = wave_reduce(acc);
    __shared__ float sh[8];
    const int wave = threadIdx.x >> 5;
    const int lane = threadIdx.x & 31;
    if (lane == 0) sh[wave] = acc;
    __syncthreads();
    if (threadIdx.x == 0) {
        float s = 0.0f;
        #pragma unroll
        for (int w = 0; w < 8; ++w) s += sh[w];    // fixed order
        partials[blockIdx.x] = s;
    }
}

// ---------------------------------------------------------------------------
// Kernel 4: single-block deterministic combine.
// out = centerSum/(2*B) + 0.5 * (gramOffDiagSum + (N^2 - N))
// ---------------------------------------------------------------------------
__global__ __launch_bounds__(256) void k_final(const float* __restrict__ gramP,
                                               const float* __restrict__ centerP,
                                               float* __restrict__ out) {
    __shared__ float shg[256];
    __shared__ float shc[256];
    float g = 0.0f, c = 0.0f;
    for (int i = threadIdx.x; i < N_TILES * N_TILES; i += 256) g += gramP[i];
    for (int i = threadIdx.x; i < CL_BLOCKS;         i += 256) c += centerP[i];
    shg[threadIdx.x] = g;
    shc[threadIdx.x] = c;
    __syncthreads();
    for (int s = 128; s > 0; s >>= 1) {
        if (threadIdx.x < s) {
            shg[threadIdx.x] += shg[threadIdx.x + s];
            shc[threadIdx.x] += shc[threadIdx.x + s];
        }
        __syncthreads();
    }
    if (threadIdx.x == 0) {
        const float center_loss = shc[0] / (2.0f * (float)BATCH);
        const float pair_count  = (float)(NUM_CLASSES * NUM_CLASSES - NUM_CLASSES); // 999000
        const float island_loss = shg[0] + pair_count;
        out[0] = center_loss + 0.5f * island_loss;
    }
}

// ---------------------------------------------------------------------------
// Launch. Inputs (setup_inputs order): label (int32 [32768]),
// feat (f32 [32768*512]), centers (f32 [1000*512]). Output: 1 float.
// Workspace layout:
//   [0, 2064384)           cn  : float[1008][512]
//   [2064384, +16384)      gram partials : float[4096] (3969 used)
//   [2080768, +16384)      center partials : float[4096]
// ---------------------------------------------------------------------------
extern "C" void kernel_launch(void* const* d_in, const int* in_sizes, int n_in,
                              void* d_out, int out_size, void* d_ws, size_t ws_size,
                              hipStream_t stream) {
    const int*   label   = (const int*)d_in[0];
    const float* feat    = (const float*)d_in[1];
    const float* centers = (const float*)d_in[2];
    float*       out     = (float*)d_out;

    char* ws = (char*)d_ws;
    float* cn       = (float*)(ws);
    float* gramP    = (float*)(ws + (size_t)N_PAD * FEAT_DIM * sizeof(float));
    float* centerP  = gramP + 4096;

    k_normalize<<<N_PAD / 8, 256, 0, stream>>>(centers, cn);
    k_gram_tiles<<<dim3(N_TILES, N_TILES), 32, 0, stream>>>(cn, gramP);
    k_center_loss<<<CL_BLOCKS, 256, 0, stream>>>(feat, centers, label, centerP);
    k_final<<<1, 256, 0, stream>>>(gramP, centerP, out);
}